// STN_68195490726482
// MI455X (gfx1250) — compile-verified
//
#include <hip/hip_runtime.h>
#include <math.h>

#define Bn 32
#define Hn 224
#define Wn 224
#define Cn 3
#define HID 64
#define KTOT (Hn*Wn*Cn)        // 150528
#define KCHUNK 336
#define NKBLK (KTOT/KCHUNK)    // 448
#define KSTEPS (KCHUNK/4)      // 84

typedef __attribute__((ext_vector_type(2))) float v2f;
typedef __attribute__((ext_vector_type(8))) float v8f;

// ---------------------------------------------------------------- zero accum
__global__ void stn_zero(float* __restrict__ acc) {
    int i = blockIdx.x * blockDim.x + threadIdx.x;
    if (i < Bn * HID) acc[i] = 0.0f;
}

// ---------------------------------------------------------------- GEMM1 (WMMA f32 16x16x4, split-K)
// A: (32 x 150528) row-major = flattened inputs
// Bw: (150528 x 64) row-major = w1
// h_acc: (32 x 64) f32, accumulated with atomics across 448 K-chunks.
__global__ void stn_gemm1(const float* __restrict__ A,
                          const float* __restrict__ Bw,
                          float* __restrict__ h_acc) {
    const int lane = threadIdx.x & 31;
    const int wave = threadIdx.x >> 5;      // 0..7
    const int m0   = (wave >> 2) * 16;      // 0 or 16   (batch rows)
    const int n0   = (wave & 3) * 16;       // 0,16,32,48 (HID cols)
    const int hi   = lane >> 4;             // 0: K offsets {0,1}; 1: K offsets {2,3}
    const int lo   = lane & 15;

    const int kbase = blockIdx.x * KCHUNK;

    // A layout (16x4 f32): lane (hi,lo) -> M = lo, a[0]=K(2*hi), a[1]=K(2*hi+1)
    const float* __restrict__ aPtr = A  + (size_t)(m0 + lo) * KTOT + kbase + 2 * hi;
    // B layout (4x16 f32): lane (hi,lo) -> N = lo, b[0]=row K(2*hi), b[1]=row K(2*hi+1)
    const float* __restrict__ bPtr = Bw + (size_t)(kbase + 2 * hi) * HID + n0 + lo;

    v8f acc = {};
    #pragma unroll 4
    for (int s = 0; s < KSTEPS; ++s) {
        v2f a, b;
        a.x = aPtr[0];
        a.y = aPtr[1];
        b.x = bPtr[0];
        b.y = bPtr[HID];
        acc = __builtin_amdgcn_wmma_f32_16x16x4_f32(
            /*neg_a=*/false, a, /*neg_b=*/false, b,
            /*c_mod=*/(short)0, acc, /*reuse_a=*/false, /*reuse_b=*/false);
        aPtr += 4;
        bPtr += 4 * HID;
    }

    // C layout (16x16 f32, 8 VGPRs): lanes 0-15 -> M=j,   N=lo
    //                                lanes16-31 -> M=8+j, N=lo
    #pragma unroll
    for (int j = 0; j < 8; ++j) {
        const int row = m0 + hi * 8 + j;
        const int col = n0 + lo;
        atomicAdd(&h_acc[row * HID + col], acc[j]);
    }
}

// ---------------------------------------------------------------- head: tanh + tiny GEMM2 -> theta
// One block of 256 threads.
__global__ void stn_head(const float* __restrict__ h_acc,
                         const float* __restrict__ b1,
                         const float* __restrict__ w2,   // (64 x 6)
                         const float* __restrict__ b2,   // (6)
                         float* __restrict__ theta) {    // (32 x 6)
    __shared__ float h_s[Bn * HID];
    const int t = threadIdx.x;
    #pragma unroll
    for (int i = t; i < Bn * HID; i += 256) {
        h_s[i] = tanhf(h_acc[i] + b1[i % HID]);
    }
    __syncthreads();
    if (t < Bn * 6) {
        const int b = t / 6;
        const int j = t % 6;
        float s = b2[j];
        #pragma unroll
        for (int k = 0; k < HID; ++k) s += h_s[b * HID + k] * w2[k * 6 + j];
        theta[t] = tanhf(s);
    }
}

// ---------------------------------------------------------------- affine grid + bilinear sampling
__global__ void stn_sample(const float* __restrict__ img,
                           const float* __restrict__ theta,
                           float* __restrict__ out) {
    const int idx = blockIdx.x * blockDim.x + threadIdx.x;
    const int npix = Bn * Hn * Wn;
    if (idx >= npix) return;

    const int b  = idx / (Hn * Wn);
    const int p  = idx - b * (Hn * Wn);
    const int yi = p / Wn;
    const int xi = p - yi * Wn;

    const float xt = -1.0f + 2.0f * (float)xi / (float)(Wn - 1);
    const float yt = -1.0f + 2.0f * (float)yi / (float)(Hn - 1);

    const float* th = theta + b * 6;
    const float x_s = th[0] * xt + th[1] * yt + th[2];
    const float y_s = th[3] * xt + th[4] * yt + th[5];

    const float x = 0.5f * (x_s + 1.0f) * (float)(Wn - 1);
    const float y = 0.5f * (y_s + 1.0f) * (float)(Hn - 1);

    const int x0 = (int)floorf(x);
    const int y0 = (int)floorf(y);
    const int x1 = x0 + 1;
    const int y1 = y0 + 1;

    const int x0c = min(max(x0, 0), Wn - 1);
    const int x1c = min(max(x1, 0), Wn - 1);
    const int y0c = min(max(y0, 0), Hn - 1);
    const int y1c = min(max(y1, 0), Hn - 1);

    const float x0f = (float)x0c, x1f = (float)x1c;
    const float y0f = (float)y0c, y1f = (float)y1c;

    const float wa = (x1f - x) * (y1f - y);
    const float wb = (x1f - x) * (y - y0f);
    const float wc = (x - x0f) * (y1f - y);
    const float wd = (x - x0f) * (y - y0f);

    const float* base = img + (size_t)b * Hn * Wn * Cn;
    const float* Ia = base + ((size_t)y0c * Wn + x0c) * Cn;
    const float* Ib = base + ((size_t)y1c * Wn + x0c) * Cn;
    const float* Ic = base + ((size_t)y0c * Wn + x1c) * Cn;
    const float* Id = base + ((size_t)y1c * Wn + x1c) * Cn;

    float* o = out + (size_t)idx * Cn;
    #pragma unroll
    for (int c = 0; c < Cn; ++c) {
        o[c] = wa * Ia[c] + wb * Ib[c] + wc * Ic[c] + wd * Id[c];
    }
}

// ---------------------------------------------------------------- launch
extern "C" void kernel_launch(void* const* d_in, const int* in_sizes, int n_in,
                              void* d_out, int out_size, void* d_ws, size_t ws_size,
                              hipStream_t stream) {
    const float* inputs = (const float*)d_in[0];
    const float* w1     = (const float*)d_in[1];
    const float* b1     = (const float*)d_in[2];
    const float* w2     = (const float*)d_in[3];
    const float* b2     = (const float*)d_in[4];
    float* out = (float*)d_out;

    float* h_acc = (float*)d_ws;               // 32*64 floats
    float* theta = h_acc + Bn * HID;           // 32*6 floats

    stn_zero<<<(Bn * HID + 255) / 256, 256, 0, stream>>>(h_acc);
    stn_gemm1<<<NKBLK, 256, 0, stream>>>(inputs, w1, h_acc);
    stn_head<<<1, 256, 0, stream>>>(h_acc, b1, w2, b2, theta);

    const int npix = Bn * Hn * Wn;
    stn_sample<<<(npix + 255) / 256, 256, 0, stream>>>(inputs, theta, out);
}